// CompatibilityLayer_58643483459745
// MI455X (gfx1250) — compile-verified
//
#include <hip/hip_runtime.h>

#define CLS 32

typedef __attribute__((ext_vector_type(2))) float v2f;
typedef __attribute__((ext_vector_type(8))) float v8f;

// ---------------------------------------------------------------------------
// 1) zero the accumulation region of the workspace
// ---------------------------------------------------------------------------
__global__ void zero_kernel(float* p, int n) {
    int i = blockIdx.x * blockDim.x + threadIdx.x;
    int stride = gridDim.x * blockDim.x;
    for (; i < n; i += stride) p[i] = 0.0f;
}

// ---------------------------------------------------------------------------
// 2) degree accumulation over edges (self-loop weight 1 added in dinv kernel)
// ---------------------------------------------------------------------------
__global__ void deg_kernel(const int* __restrict__ row, const float* __restrict__ w,
                           float* __restrict__ deg, int E) {
    int e = blockIdx.x * blockDim.x + threadIdx.x;
    if (e < E) atomicAdd(&deg[row[e]], w[e]);
}

// ---------------------------------------------------------------------------
// 3) dinv = rsqrt(deg + 1)   (deg+1 > 0 always: self-loop weight is 1)
// ---------------------------------------------------------------------------
__global__ void dinv_kernel(const float* __restrict__ deg, float* __restrict__ dinv, int N) {
    int i = blockIdx.x * blockDim.x + threadIdx.x;
    if (i < N) dinv[i] = rsqrtf(deg[i] + 1.0f);
}

// ---------------------------------------------------------------------------
// 4) per-node softmax (one wave32 per node, lane = class), blend with labels
//    on masked nodes, and accumulate per-class masked counts.
// ---------------------------------------------------------------------------
__global__ void softmax_blend_kernel(const float* __restrict__ x, const float* __restrict__ y,
                                     const unsigned char* __restrict__ mask,
                                     float* __restrict__ p_eff, float* __restrict__ cnt, int N) {
    int tid = blockIdx.x * blockDim.x + threadIdx.x;
    int n = tid >> 5;
    int c = tid & 31;
    if (n >= N) return;
    float v = x[n * CLS + c];
    float mx = v;
    #pragma unroll
    for (int off = 16; off; off >>= 1) mx = fmaxf(mx, __shfl_xor(mx, off, 32));
    float e = __expf(v - mx);
    float s = e;
    #pragma unroll
    for (int off = 16; off; off >>= 1) s += __shfl_xor(s, off, 32);
    float p = e / s;
    bool msk = mask[n] != 0;
    float yv = y[n * CLS + c];
    float pe = msk ? yv : p;
    p_eff[n * CLS + c] = pe;
    float ym = msk ? yv : 0.0f;
    if (ym != 0.0f) atomicAdd(&cnt[c], ym);
}

// ---------------------------------------------------------------------------
// 5) SpMM edges: nodeH[row] += dinv[row]*w*dinv[col] * p_eff[col]
//    8 lanes per edge, float4 per lane. p_eff/nodeH (25.6 MB) stay in L2.
// ---------------------------------------------------------------------------
__global__ void spmm_edge_kernel(const int* __restrict__ ei, const float* __restrict__ w,
                                 const float* __restrict__ dinv, const float* __restrict__ p_eff,
                                 float* __restrict__ nodeH, int E) {
    int tid = blockIdx.x * blockDim.x + threadIdx.x;
    int e = tid >> 3;
    int g = tid & 7;
    if (e >= E) return;
    int r = ei[e];
    int c = ei[E + e];
    float val = dinv[r] * w[e] * dinv[c];
    float4 p4 = ((const float4*)(p_eff + c * CLS))[g];
    float* dst = nodeH + r * CLS + g * 4;
    atomicAdd(dst + 0, val * p4.x);
    atomicAdd(dst + 1, val * p4.y);
    atomicAdd(dst + 2, val * p4.z);
    atomicAdd(dst + 3, val * p4.w);
}

// ---------------------------------------------------------------------------
// 6) SpMM self loops: nodeH[i] += dinv[i]^2 * p_eff[i] (exclusive per node,
//    sequenced after edge kernel -> plain read-modify-write)
// ---------------------------------------------------------------------------
__global__ void spmm_self_kernel(const float* __restrict__ dinv, const float* __restrict__ p_eff,
                                 float* __restrict__ nodeH, int N) {
    int tid = blockIdx.x * blockDim.x + threadIdx.x;
    int n = tid >> 3;
    int g = tid & 7;
    if (n >= N) return;
    float v = dinv[n];
    v = v * v;
    float4 p4 = ((const float4*)(p_eff + n * CLS))[g];
    float* dst = nodeH + n * CLS + g * 4;
    dst[0] += v * p4.x;
    dst[1] += v * p4.y;
    dst[2] += v * p4.z;
    dst[3] += v * p4.w;
}

// ---------------------------------------------------------------------------
// 7) Hacc = ym^T @ nodeH via V_WMMA_F32_16X16X4_F32.
//    Each wave owns four 16x16 f32 tiles (2x2 over the 32x32 output) and
//    reduces K = nodes in chunks of 4; partials merged with atomics.
//    Wave index / stride are forced into SGPRs via readfirstlane so the
//    K-loop condition is scalar (s_cmp/s_cbranch): EXEC stays all-1s across
//    the v_wmma block, per the ISA requirement, with no per-iteration
//    exec-mask manipulation.
//    A-layout (16x4 f32): lane L: m=L&15, VGPR0<->K=2*(L>>4), VGPR1<->K=+1.
//    B-layout (4x16 f32): lane L: n=L&15, mirrored K mapping.
//    C/D (16x16 f32): lane L: n=L&15, VGPR r <-> M=(L>>4)*8+r.
// ---------------------------------------------------------------------------
__global__ void wmma_H_kernel(const float* __restrict__ y, const unsigned char* __restrict__ mask,
                              const float* __restrict__ nodeH, float* __restrict__ Hacc, int N) {
    int lane = threadIdx.x & 31;
    int waveInBlock = threadIdx.x >> 5;
    int wavesPerBlock = blockDim.x >> 5;
    // wave-uniform values -> force to SGPRs so the loop branch is scalar
    int wave   = __builtin_amdgcn_readfirstlane(blockIdx.x * wavesPerBlock + waveInBlock);
    int nwaves = __builtin_amdgcn_readfirstlane(gridDim.x * wavesPerBlock);
    int kstep  = nwaves * 4;

    int m = lane & 15;
    int khalf = lane >> 4;

    v8f c00 = {}; v8f c01 = {}; v8f c10 = {}; v8f c11 = {};

    for (int k0 = wave * 4; k0 < N; k0 += kstep) {   // scalar loop bound
        int ka = k0 + 2 * khalf;
        int kb = ka + 1;
        float ma = (mask[ka] != 0) ? 1.0f : 0.0f;
        float mb = (mask[kb] != 0) ? 1.0f : 0.0f;
        v2f aLo, aHi, bLo, bHi;
        aLo[0] = y[ka * CLS + m] * ma;
        aLo[1] = y[kb * CLS + m] * mb;
        aHi[0] = y[ka * CLS + 16 + m] * ma;
        aHi[1] = y[kb * CLS + 16 + m] * mb;
        bLo[0] = nodeH[ka * CLS + m];
        bLo[1] = nodeH[kb * CLS + m];
        bHi[0] = nodeH[ka * CLS + 16 + m];
        bHi[1] = nodeH[kb * CLS + 16 + m];
        c00 = __builtin_amdgcn_wmma_f32_16x16x4_f32(false, aLo, false, bLo, (short)0, c00, false, false);
        c01 = __builtin_amdgcn_wmma_f32_16x16x4_f32(false, aLo, false, bHi, (short)0, c01, false, false);
        c10 = __builtin_amdgcn_wmma_f32_16x16x4_f32(false, aHi, false, bLo, (short)0, c10, false, false);
        c11 = __builtin_amdgcn_wmma_f32_16x16x4_f32(false, aHi, false, bHi, (short)0, c11, false, false);
    }

    int nOut = lane & 15;
    int mBase = (lane >> 4) * 8;
    #pragma unroll
    for (int r = 0; r < 8; ++r) {
        atomicAdd(&Hacc[(mBase + r) * CLS + nOut],            c00[r]);
        atomicAdd(&Hacc[(mBase + r) * CLS + 16 + nOut],       c01[r]);
        atomicAdd(&Hacc[(16 + mBase + r) * CLS + nOut],       c10[r]);
        atomicAdd(&Hacc[(16 + mBase + r) * CLS + 16 + nOut],  c11[r]);
    }
}

// ---------------------------------------------------------------------------
// 8) finalize: H = Hacc/cnt, NaN fixups, 3000-iteration double-stochastic
//    normalization. One wave32; lane j owns column j in 32 registers.
//    Column sums are lane-local; row sums are 5-stage shfl_xor butterflies.
// ---------------------------------------------------------------------------
__global__ void finalize_kernel(const float* __restrict__ Hacc, const float* __restrict__ cnt,
                                float* __restrict__ out) {
    int j = threadIdx.x;   // 0..31, column owner
    float Hl[32];
    float cj = cnt[j];

    #pragma unroll
    for (int i = 0; i < 32; ++i) Hl[i] = Hacc[i * CLS + j] / cnt[i];

    // fixup 1: NaN -> transposed entry (H[j][i] = row j of Hacc / cnt[j])
    #pragma unroll
    for (int i = 0; i < 32; ++i) {
        float ht = Hacc[j * CLS + i] / cj;
        if (Hl[i] != Hl[i]) Hl[i] = ht;
    }

    // fixup 2: remaining NaNs -> (1 - rowsum(H0)) / nan_count(row)
    #pragma unroll
    for (int i = 0; i < 32; ++i) {
        bool isn = (Hl[i] != Hl[i]);
        float h0 = isn ? 0.0f : Hl[i];
        float rs = h0;
        float nc = isn ? 1.0f : 0.0f;
        #pragma unroll
        for (int off = 16; off; off >>= 1) {
            rs += __shfl_xor(rs, off, 32);
            nc += __shfl_xor(nc, off, 32);
        }
        Hl[i] = isn ? (1.0f - rs) / nc : h0;
    }

    // makeDoubleStochasticH: col-normalize then row-normalize, up to 3000 iters
    float delta = 1e30f;
    int it = 0;
    #pragma unroll 1
    while (delta >= 1e-12f && it < 3000) {
        float cs = 0.0f;
        #pragma unroll
        for (int i = 0; i < 32; ++i) cs += Hl[i];
        #pragma unroll
        for (int i = 0; i < 32; ++i) Hl[i] /= cs;
        float d = 0.0f;
        #pragma unroll
        for (int i = 0; i < 32; ++i) {
            float r = Hl[i];
            #pragma unroll
            for (int off = 16; off; off >>= 1) r += __shfl_xor(r, off, 32);
            Hl[i] /= r;
            d += fabsf(Hl[i]);
        }
        #pragma unroll
        for (int off = 16; off; off >>= 1) d += __shfl_xor(d, off, 32);
        delta = d;
        ++it;
    }

    #pragma unroll
    for (int i = 0; i < 32; ++i) out[i * CLS + j] = Hl[i];
}

// ---------------------------------------------------------------------------
extern "C" void kernel_launch(void* const* d_in, const int* in_sizes, int n_in,
                              void* d_out, int out_size, void* d_ws, size_t ws_size,
                              hipStream_t stream) {
    const int* edge_index = (const int*)d_in[0];            // (2, E) int32
    const float* edge_weight = (const float*)d_in[1];       // (E,)
    const float* inputs = (const float*)d_in[2];            // (N, 32)
    const float* y = (const float*)d_in[3];                 // (N, 32)
    const unsigned char* mask = (const unsigned char*)d_in[4]; // (N,) bool

    const int E = in_sizes[1];
    const int N = in_sizes[4];

    float* ws = (float*)d_ws;
    // layout: [deg:N][cnt:32][Hacc:1024][nodeH:N*32] (zeroed) | [dinv:N][p_eff:N*32]
    float* deg   = ws;
    float* cnt   = ws + N;
    float* Hacc  = ws + N + 32;
    float* nodeH = ws + N + 32 + 1024;
    float* dinv  = nodeH + N * CLS;
    float* p_eff = dinv + N;
    const int zeroCount = N + 32 + 1024 + N * CLS;

    const int B = 256;
    zero_kernel<<<2048, B, 0, stream>>>(ws, zeroCount);
    deg_kernel<<<(E + B - 1) / B, B, 0, stream>>>(edge_index, edge_weight, deg, E);
    dinv_kernel<<<(N + B - 1) / B, B, 0, stream>>>(deg, dinv, N);
    softmax_blend_kernel<<<(N * CLS + B - 1) / B, B, 0, stream>>>(inputs, y, mask, p_eff, cnt, N);
    spmm_edge_kernel<<<(E * 8 + B - 1) / B, B, 0, stream>>>(edge_index, edge_weight, dinv, p_eff, nodeH, E);
    spmm_self_kernel<<<(N * 8 + B - 1) / B, B, 0, stream>>>(dinv, p_eff, nodeH, N);
    wmma_H_kernel<<<128, B, 0, stream>>>(y, mask, nodeH, Hacc, N);
    finalize_kernel<<<1, 32, 0, stream>>>(Hacc, cnt, (float*)d_out);
}